// GLSMIFTDescriptor_82952998355300
// MI455X (gfx1250) — compile-verified
//
#include <hip/hip_runtime.h>
#include <stdint.h>

// Problem constants (from reference)
#define POINT_FLOATS (24 * 18 * 96)  // 41472 floats per point
#define ROW 1728                     // floats per filter plane (18 parts * 96 pix)
#define PIX 96
#define TILE_W 192                   // floats of each filter row per tile
#define NTILES 9                     // 1728 / 192
#define TILE_FLOATS (24 * TILE_W)    // 4608 floats = 18432 B per tile
#define BLK 192                      // 6 wave32s; 192 chunks (4 sigma x 48 float4) per tile

typedef uint32_t u32;
typedef unsigned int v4u __attribute__((ext_vector_type(4)));
typedef int v8i __attribute__((ext_vector_type(8)));
typedef int v4i __attribute__((ext_vector_type(4)));

// Issue one TDM 2D tile load: 24 rows x TILE_W floats, row stride ROW floats,
// from global (tile start) into LDS at byte offset lds_off.
__device__ __forceinline__ void tdm_load_tile(const float* gtile, u32 lds_off) {
  uint64_t ga = (uint64_t)(uintptr_t)gtile;
  v4u g0;
  g0[0] = 1u;                                            // count=1, no gather
  g0[1] = lds_off;                                       // lds_addr [63:32]
  g0[2] = (u32)ga;                                       // global_addr[31:0]
  g0[3] = ((u32)(ga >> 32) & 0x01FFFFFFu) | (2u << 30);  // addr[56:32] | type=2
  v8i g1;
  g1[0] = 0x00020000;                    // workgroup_mask=0, data_size=2 (4B)
  g1[1] = (int)(((u32)ROW & 0xFFFFu) << 16);   // tensor_dim0[15:0] @ bits 63:48
  g1[2] = (int)(24u << 16);              // tensor_dim0[31:16]=0, tensor_dim1=24
  g1[3] = (int)(((u32)TILE_W) << 16);    // tensor_dim1[31:16]=0, tile_dim0=192
  g1[4] = 24;                            // tile_dim1=24, tile_dim2=0
  g1[5] = ROW;                           // tensor_dim0_stride[31:0] = 1728
  g1[6] = 0;                             // stride hi, tensor_dim1_stride lo
  g1[7] = 0;
  v4i z4 = {0, 0, 0, 0};                 // groups 2/3 unused for 2D tile
  v8i z8 = {0, 0, 0, 0, 0, 0, 0, 0};     // trailing group (6-arg toolchain form)
  __builtin_amdgcn_tensor_load_to_lds(g0, g1, z4, z4, z8, 0);
}

// Decode packed histogram: per part, 2 words x 3 fields of 10 bits (max 384 < 1024)
__device__ __forceinline__ u32 hget(const u32* hist, int p, int a) {
  return (hist[2 * p + (a >= 3 ? 1 : 0)] >> (10 * (a % 3))) & 0x3FFu;
}

__global__ void __launch_bounds__(BLK) glsmift_kernel(const float* __restrict__ patches,
                                                      float* __restrict__ out, int npts) {
  __shared__ __align__(16) float buf[2][TILE_FLOATS];
  __shared__ u32 hist[36];
  __shared__ u32 prim[2];
  __shared__ float vals[108];
  __shared__ float red;

  const int n = blockIdx.x;
  if (n >= npts) return;
  const float* pbase = patches + (size_t)n * POINT_FLOATS;
  const int tid = threadIdx.x;

  if (tid < 36) hist[tid] = 0u;

  const bool is_w0 = (tid < 32);  // wave 0 (wave32) drives the TDM
  const u32 lds0 = (u32)(uintptr_t)&buf[0][0];
  const u32 lds1 = (u32)(uintptr_t)&buf[1][0];

  if (is_w0) tdm_load_tile(pbase, lds0);

  const int s = tid / 48;  // sigma 0..3
  const int q = tid % 48;  // float4 column index within TILE_W

  for (int t = 0; t < NTILES; ++t) {
    if (is_w0) {
      if (t + 1 < NTILES) {
        tdm_load_tile(pbase + (t + 1) * TILE_W, ((t + 1) & 1) ? lds1 : lds0);
        __builtin_amdgcn_s_wait_tensorcnt((short)1);  // tile t complete, t+1 in flight
      } else {
        __builtin_amdgcn_s_wait_tensorcnt((short)0);
      }
    }
    __syncthreads();

    const float* b = buf[t & 1];
    float vv[6][4];
#pragma unroll
    for (int a = 0; a < 6; ++a) {
      const float4 f4 = *reinterpret_cast<const float4*>(&b[(a * 4 + s) * TILE_W + 4 * q]);
      vv[a][0] = f4.x; vv[a][1] = f4.y; vv[a][2] = f4.z; vv[a][3] = f4.w;
    }
    u32 wlo = 0, whi = 0;
#pragma unroll
    for (int e = 0; e < 4; ++e) {
      float best = vv[0][e];
      int bi = 0;
#pragma unroll
      for (int a = 1; a < 6; ++a) {  // strict '>' => first-occurrence like jnp.argmax
        float x = vv[a][e];
        if (x > best) { best = x; bi = a; }
      }
      if (bi < 3) wlo += 1u << (10 * bi);
      else        whi += 1u << (10 * (bi - 3));
    }
    const int y = t * TILE_W + 4 * q;  // position in flattened (part,pix)
    const int p = y / PIX;             // part 0..17 (float4 never crosses a part)
    atomicAdd(&hist[2 * p + 0], wlo);
    atomicAdd(&hist[2 * p + 1], whi);
    __syncthreads();
  }

  // ---- primary direction & angular part (tiny; serial on thread 0) ----
  if (tid == 0) {
    u32 dc[6];
    for (int a = 0; a < 6; ++a) {
      u32 c = 0;
      for (int p = 0; p < 18; ++p) c += hget(hist, p, a);
      dc[a] = c;
    }
    int pd = 0;
    for (int a = 1; a < 6; ++a) if (dc[a] > dc[pd]) pd = a;
    u32 ac[6];
    for (int ap = 0; ap < 6; ++ap)
      ac[ap] = hget(hist, ap * 3 + 0, pd) + hget(hist, ap * 3 + 1, pd) + hget(hist, ap * 3 + 2, pd);
    int pap = 0;
    for (int ap = 1; ap < 6; ++ap) if (ac[ap] > ac[pap]) pap = ap;
    prim[0] = (u32)pd;
    prim[1] = (u32)pap;
  }
  __syncthreads();

  // ---- shift-gather + /384 ----
  if (tid < 108) {
    const int j = tid / 18, k = (tid % 18) / 6, l = tid % 6;
    const int pp = (int)((j + prim[1]) % 6u) * 3 + k;
    const int aa = (int)((l + prim[0]) % 6u);
    vals[tid] = (float)hget(hist, pp, aa) / 384.0f;
  }
  __syncthreads();
  // ---- RootSIFT: L1 -> sqrt -> L2 ----
  if (tid == 0) {
    float s1 = 0.f;
    for (int i = 0; i < 108; ++i) s1 += fabsf(vals[i]);
    red = fmaxf(s1, 1e-12f);
  }
  __syncthreads();
  if (tid < 108) vals[tid] = sqrtf(vals[tid] / red);
  __syncthreads();
  if (tid == 0) {
    float s2 = 0.f;
    for (int i = 0; i < 108; ++i) s2 += vals[i] * vals[i];
    red = fmaxf(sqrtf(s2), 1e-12f);
  }
  __syncthreads();
  if (tid < 108) out[(size_t)n * 108 + tid] = vals[tid] / red;
}

extern "C" void kernel_launch(void* const* d_in, const int* in_sizes, int n_in,
                              void* d_out, int out_size, void* d_ws, size_t ws_size,
                              hipStream_t stream) {
  const float* patches = (const float*)d_in[0];
  float* out = (float*)d_out;
  const int npts = in_sizes[0] / POINT_FLOATS;  // B*N = 5000
  glsmift_kernel<<<dim3(npts), dim3(BLK), 0, stream>>>(patches, out, npts);
}